// EdgeConv_21646635172271
// MI455X (gfx1250) — compile-verified
//
#include <hip/hip_runtime.h>
#include <hip/hip_bf16.h>
#include <math.h>

typedef float v2f __attribute__((ext_vector_type(2)));
typedef float v8f __attribute__((ext_vector_type(8)));

#define NUM_NODES 32768   // B*N = 32*1024
#define DIM 64            // D_IN == D_OUT
#define NUM_EDGES 1048576

// ---------------------------------------------------------------------------
// Projection: xt = x @ Wt^T ; xp = x @ Wp^T + (bt + bp)
// One wave computes one 16(M:nodes) x 16(N:outdims) tile using
// V_WMMA_F32_16X16X4_F32, 16 K-steps of 4, two accumulators sharing A loads.
// Block = 128 threads = 4 waves, each wave a distinct M tile. EXEC all-ones.
// ---------------------------------------------------------------------------
__global__ __launch_bounds__(128) void proj_wmma(
    const float* __restrict__ x,   // [NUM_NODES, DIM]
    const float* __restrict__ Wt,  // [DIM, DIM]
    const float* __restrict__ Wp,  // [DIM, DIM]
    const float* __restrict__ bt,  // [DIM]
    const float* __restrict__ bp,  // [DIM]
    float* __restrict__ xt,        // [NUM_NODES, DIM]
    float* __restrict__ xp)        // [NUM_NODES, DIM]
{
    const int lane  = threadIdx.x & 31;
    const int wave  = threadIdx.x >> 5;
    const int mtile = blockIdx.x * 4 + wave;   // 2048 M tiles
    const int otile = blockIdx.y;              // 4 N tiles
    const int m0 = mtile * 16;
    const int o0 = otile * 16;

    // A layout (16x4 f32): lane L -> M = L%16, K-pair base = (L/16)*2
    // B layout (4x16 f32): lane L -> N = L%16, K-pair base = (L/16)*2
    const int mrow  = m0 + (lane & 15);
    const int orow  = o0 + (lane & 15);
    const int khalf = (lane >> 4) * 2;

    v8f accT = {};
    v8f accP = {};

    #pragma unroll
    for (int k = 0; k < DIM; k += 4) {
        v2f a, bT, bP;
        a.x  = x [mrow * DIM + k + khalf + 0];
        a.y  = x [mrow * DIM + k + khalf + 1];
        // B[k][n] = W^T[k][n] = W[n][k]
        bT.x = Wt[orow * DIM + k + khalf + 0];
        bT.y = Wt[orow * DIM + k + khalf + 1];
        bP.x = Wp[orow * DIM + k + khalf + 0];
        bP.y = Wp[orow * DIM + k + khalf + 1];
        accT = __builtin_amdgcn_wmma_f32_16x16x4_f32(
            false, a, false, bT, (short)0, accT, false, false);
        accP = __builtin_amdgcn_wmma_f32_16x16x4_f32(
            false, a, false, bP, (short)0, accP, false, false);
    }

    // Fold edge-invariant bias (bt+bp) into xp (cancels out of xt[dst]-xt[src]).
    const float bsum = bt[orow] + bp[orow];

    // D layout: VGPR v in lane L -> (M = v + 8*(L/16), N = L%16)
    const int mb = m0 + 8 * (lane >> 4);
    const int oc = o0 + (lane & 15);
    #pragma unroll
    for (int v = 0; v < 8; ++v) {
        const int m = mb + v;
        xt[m * DIM + oc] = accT[v];
        xp[m * DIM + oc] = accP[v] + bsum;
    }
}

// ---------------------------------------------------------------------------
// Init output to -inf (segment_max identity).
// ---------------------------------------------------------------------------
__global__ __launch_bounds__(256) void init_out(float* __restrict__ out, int n)
{
    int i = blockIdx.x * 256 + threadIdx.x;
    if (i < n) out[i] = -INFINITY;
}

// ---------------------------------------------------------------------------
// Native L2-side fp32 max atomic (no return -> STOREcnt, fire-and-forget).
// Guarantees GLOBAL_ATOMIC_MAX_NUM_F32 instead of a header-dependent CAS loop.
// ---------------------------------------------------------------------------
__device__ __forceinline__ void atomic_max_f32(float* addr, float val)
{
    asm volatile("global_atomic_max_num_f32 %0, %1, off"
                 :
                 : "v"(addr), "v"(val)
                 : "memory");
}

// ---------------------------------------------------------------------------
// Edge phase: one wave per edge (tid>>5 is wave-uniform under wave32 so
// src/dst loads scalarize); each lane handles a float2 of the 64-dim row.
// e = xt[dst] - xt[src] + xp'[src]  ;  out[dst] = max(out[dst], e) atomically.
// ---------------------------------------------------------------------------
__global__ __launch_bounds__(256) void edge_max(
    const float* __restrict__ xt,
    const float* __restrict__ xp,
    const int*   __restrict__ src,
    const int*   __restrict__ dst,
    float* __restrict__ out)
{
    const long long tid = (long long)blockIdx.x * 256 + threadIdx.x;
    const int e = (int)(tid >> 5);
    if (e >= NUM_EDGES) return;
    const int p = (int)(tid & 31) * 2;   // dim pair: 0,2,...,62

    const int s = src[e];
    const int d = dst[e];

    const float2 a = *(const float2*)(xt + (size_t)d * DIM + p);
    const float2 b = *(const float2*)(xt + (size_t)s * DIM + p);
    const float2 c = *(const float2*)(xp + (size_t)s * DIM + p);

    const float v0 = a.x - b.x + c.x;
    const float v1 = a.y - b.y + c.y;

    float* o = out + (size_t)d * DIM + p;
    atomic_max_f32(o + 0, v0);
    atomic_max_f32(o + 1, v1);
}

// ---------------------------------------------------------------------------
// Nodes with no incoming edges stay -inf -> replace non-finite with 0.
// ---------------------------------------------------------------------------
__global__ __launch_bounds__(256) void finalize_out(float* __restrict__ out, int n)
{
    int i = blockIdx.x * 256 + threadIdx.x;
    if (i < n) {
        float v = out[i];
        out[i] = isfinite(v) ? v : 0.0f;
    }
}

extern "C" void kernel_launch(void* const* d_in, const int* in_sizes, int n_in,
                              void* d_out, int out_size, void* d_ws, size_t ws_size,
                              hipStream_t stream)
{
    // setup_inputs order: h, W_theta, b_theta, W_phi, b_phi, src, dst
    const float* h   = (const float*)d_in[0];
    const float* Wt  = (const float*)d_in[1];
    const float* bt  = (const float*)d_in[2];
    const float* Wp  = (const float*)d_in[3];
    const float* bp  = (const float*)d_in[4];
    const int*   src = (const int*)d_in[5];
    const int*   dst = (const int*)d_in[6];
    float* out = (float*)d_out;

    float* xt = (float*)d_ws;                       // 8 MB
    float* xp = xt + (size_t)NUM_NODES * DIM;       // 8 MB

    const int n_out = NUM_NODES * DIM;              // 2,097,152

    // 1) out = -inf
    init_out<<<(n_out + 255) / 256, 256, 0, stream>>>(out, n_out);

    // 2) node-level projections via fp32 WMMA (2048 M-tiles x 4 N-tiles)
    dim3 pgrid(NUM_NODES / 16 / 4, DIM / 16, 1);    // (512, 4)
    proj_wmma<<<pgrid, 128, 0, stream>>>(h, Wt, Wp, bt, bp, xt, xp);

    // 3) edge gather + atomic max scatter: 32 lanes per edge
    const long long work = (long long)NUM_EDGES * 32;
    edge_max<<<(unsigned)((work + 255) / 256), 256, 0, stream>>>(xt, xp, src, dst, out);

    // 4) -inf -> 0
    finalize_out<<<(n_out + 255) / 256, 256, 0, stream>>>(out, n_out);
}